// Reorg_28415503631135
// MI455X (gfx1250) — compile-verified
//
#include <hip/hip_runtime.h>
#include <hip/hip_bf16.h>

// Reorg / space-to-depth, stride 2.
// in : (B=8, C=512, H=128, W=128) f32
// out: (B=8, 4C=2048, Hh=64, Ww=64) f32, oc = hs*2C + ws*C + c
//
// Pure data movement: 256 MB in + 256 MB out => ~22 us floor at 23.3 TB/s.
// Each thread owns 32 B of one input row (8 floats = 4 ww-pairs, both ws
// phases), deinterleaves in registers via shufflevector, and writes one
// 16 B float4 into the even-phase output row and one into the odd-phase row.
// All VMEM ops are 128-bit, non-temporal, and fully lane-coalesced.

typedef float v4f __attribute__((ext_vector_type(4)));

namespace {
constexpr int kC  = 512;
constexpr int kH  = 128;
constexpr int kW  = 128;
constexpr int kHh = kH / 2;   // 64
constexpr int kWw = kW / 2;   // 64

constexpr int kW8        = kW / 8;               // 8-float groups per input row = 16
constexpr int kOddSkipF4 = (kC * kHh * kWw) / 4; // even->odd row distance, v4f units
}

__global__ __launch_bounds__(256) void Reorg_28415503631135_kernel(
    const v4f* __restrict__ in, v4f* __restrict__ out, int n8) {
  const int t = blockIdx.x * blockDim.x + threadIdx.x;
  if (t >= n8) return;

  // Decompose t -> (b, c, h, i) ; i = 8-float group index within the W row.
  const int i   = t & (kW8 - 1);      // 0..15
  const int row = t >> 4;             // (b*C + c)*H + h
  const int h   = row & (kH - 1);     // 0..127
  const int bc  = row >> 7;           // b*C + c
  const int hs  = h & 1;
  const int hh  = h >> 1;
  const int c   = bc & (kC - 1);
  const int b   = bc >> 9;

  // Two streaming 128-bit loads: w = 8i..8i+3 and 8i+4..8i+7.
  const v4f v0 = __builtin_nontemporal_load(&in[2 * t]);
  const v4f v1 = __builtin_nontemporal_load(&in[2 * t + 1]);

  // Register deinterleave: even (ws=0) -> ww = 4i..4i+3, odd (ws=1) likewise.
  const v4f even = __builtin_shufflevector(v0, v1, 0, 2, 4, 6);
  const v4f odd  = __builtin_shufflevector(v0, v1, 1, 3, 5, 7);

  // Even output channel: oc0 = hs*2C + c (ws = 0). Odd adds C channels.
  const int oc0 = (hs << 10) + c;                       // 2C = 1024
  // v4f-granular output index: (((b*4C + oc0)*Hh + hh)*Ww + 4i) / 4
  const int idx_e = ((b * (4 * kC) + oc0) * kHh + hh) * (kWw / 4) + i;
  const int idx_o = idx_e + kOddSkipF4;

  __builtin_nontemporal_store(even, &out[idx_e]);
  __builtin_nontemporal_store(odd,  &out[idx_o]);
}

extern "C" void kernel_launch(void* const* d_in, const int* in_sizes, int n_in,
                              void* d_out, int out_size, void* d_ws, size_t ws_size,
                              hipStream_t stream) {
  const v4f* in = (const v4f*)d_in[0];
  v4f* out = (v4f*)d_out;

  const int n8 = in_sizes[0] / 8;                  // 8,388,608 8-float groups
  const int threads = 256;                         // 8 wave32s per block
  const int blocks = (n8 + threads - 1) / threads; // 32,768
  Reorg_28415503631135_kernel<<<blocks, threads, 0, stream>>>(in, out, n8);
}